// TemporalShift_12326556139928
// MI455X (gfx1250) — compile-verified
//
#include <hip/hip_runtime.h>
#include <stdint.h>

// Temporal Shift Module (residual mode == plain shift), pure streaming copy.
// x: [B=8, T=16, C=256, H=56, W=56] f32.
//   c in [0,16)  : out[b,t,c] = x[b,t-1,c]   (0 at t==0)
//   c in [16,32) : out[b,t,c] = x[b,t+1,c]   (0 at t==T-1)
//   c >= 32      : out = x
// Memory-bound: ~822 MB traffic -> ~35 us at 23.3 TB/s. Zero FLOPs, so WMMA
// is the wrong unit; the chip-specific path is the gfx1250 async global<->LDS
// data mover (ASYNCcnt) + b128 + non-temporal hints (no reuse of 411 MB
// streams in the 192 MB L2).

typedef __attribute__((ext_vector_type(4))) float v4f;
typedef __attribute__((ext_vector_type(4))) int   v4i;

#define GLOBAL_AS __attribute__((address_space(1)))
#define LDS_AS    __attribute__((address_space(3)))

typedef GLOBAL_AS v4i* gp4;   // async builtins take non-const AS1 int4*
typedef LDS_AS    v4i* lp4;   // AS3 int4*

#if defined(__has_builtin)
#if __has_builtin(__builtin_amdgcn_global_load_async_to_lds_b128) && \
    __has_builtin(__builtin_amdgcn_global_store_async_from_lds_b128) && \
    __has_builtin(__builtin_amdgcn_s_wait_asynccnt)
#define TSM_ASYNC 1
#endif
#endif
#ifndef TSM_ASYNC
#define TSM_ASYNC 0
#endif

namespace {
constexpr unsigned Tdim   = 16;
constexpr unsigned Cdim   = 256;
constexpr unsigned HW     = 56 * 56;          // 3136 floats per (b,t,c) plane
constexpr unsigned PLANE4 = HW / 4;           // 784 float4 per plane (784 % 4 == 0)
constexpr unsigned NFWD   = 16;               // channels shifted from t-1
constexpr unsigned NSHIFT = 32;               // channels 16..31 shifted from t+1
constexpr int      DT4    = (int)(Cdim * PLANE4); // float4 stride of t +/- 1 at fixed c
constexpr unsigned CHUNK  = 4;                // float4 per thread (64 B)
constexpr unsigned BLOCK  = 256;              // 8 waves (wave32)
}

__global__ __launch_bounds__(BLOCK) void tsm_shift_kernel(
    const v4f* __restrict__ x, v4f* __restrict__ out, unsigned nthreads) {
  unsigned tid = blockIdx.x * BLOCK + threadIdx.x;
  if (tid >= nthreads) return;              // grid is exact; EXEC stays full

  // First float4 of this thread's 64B group; group never crosses a plane.
  unsigned base  = tid * CHUNK;
  unsigned plane = base / PLANE4;           // constant-divisor -> mul-hi
  unsigned c     = plane % Cdim;
  unsigned t     = (plane / Cdim) % Tdim;

  int  src  = (int)base;
  bool zero = false;
  if (c < NFWD) {                           // take from t-1
    zero = (t == 0);
    src -= DT4;
  } else if (c < NSHIFT) {                  // take from t+1
    zero = (t == Tdim - 1);
    src += DT4;
  }

#if TSM_ASYNC
  __shared__ v4f stage[BLOCK * CHUNK];      // 16 KB/block -> 20 blocks/WGP in 320 KB
  if (zero) {
    v4f z = (v4f)(0.0f);
#pragma unroll
    for (unsigned i = 0; i < CHUNK; ++i)
      __builtin_nontemporal_store(z, out + base + i);   // global_store_b128 th:NT
  } else {
    // Raw 32-bit LDS offset: flat aperture keeps the LDS byte offset in addr[31:0].
    unsigned lds_off = (unsigned)(uintptr_t)&stage[threadIdx.x * CHUNK];
#pragma unroll
    for (unsigned i = 0; i < CHUNK; ++i) {
      __builtin_amdgcn_global_load_async_to_lds_b128(
          (gp4)(uintptr_t)(x + src + (int)i),
          (lp4)(lds_off + 16u * i), 0, 0);              // ASYNCcnt++
    }
    __builtin_amdgcn_s_wait_asynccnt(0);                // loads landed in LDS
#pragma unroll
    for (unsigned i = 0; i < CHUNK; ++i) {
      __builtin_amdgcn_global_store_async_from_lds_b128(
          (gp4)(uintptr_t)(out + base + i),
          (lp4)(lds_off + 16u * i), 0, 0);              // ASYNCcnt++
    }
  }
#else
  // Fallback: direct VGPR pass-through, b128 non-temporal both ways.
#pragma unroll
  for (unsigned i = 0; i < CHUNK; ++i) {
    v4f v = (v4f)(0.0f);
    if (!zero) v = __builtin_nontemporal_load(x + src + (int)i);
    __builtin_nontemporal_store(v, out + base + i);
  }
#endif
}

extern "C" void kernel_launch(void* const* d_in, const int* in_sizes, int n_in,
                              void* d_out, int out_size, void* d_ws, size_t ws_size,
                              hipStream_t stream) {
  (void)n_in; (void)out_size; (void)d_ws; (void)ws_size;
  const v4f* x   = (const v4f*)d_in[0];
  v4f*       out = (v4f*)d_out;

  const unsigned total4   = (unsigned)in_sizes[0] / 4u;  // 25,690,112 float4
  const unsigned nthreads = total4 / CHUNK;              // 6,422,528
  const unsigned grid     = (nthreads + BLOCK - 1) / BLOCK; // 25,088 exact
  tsm_shift_kernel<<<grid, BLOCK, 0, stream>>>(x, out, nthreads);
}